// LSTM_77025943486858
// MI455X (gfx1250) — compile-verified
//
#include <hip/hip_runtime.h>
#include <stdint.h>

typedef __attribute__((ext_vector_type(2))) float        v2f;
typedef __attribute__((ext_vector_type(8))) float        v8f;
typedef __attribute__((ext_vector_type(4))) unsigned int u32x4;
typedef __attribute__((ext_vector_type(8))) int          i32x8;
typedef __attribute__((ext_vector_type(4))) int          i32x4;

#define TC 16   // timesteps staged per TDM chunk
#define IN 8    // input dim
#define NG 24   // 4*H gate rows
#define HD 6    // hidden dim

__device__ __forceinline__ float fast_sigmoid(float x) {
  float e = __expf(-x);
  return __builtin_amdgcn_rcpf(1.0f + e);
}
__device__ __forceinline__ float fast_tanh(float x) {
  float e = __expf(-2.0f * x);
  return (1.0f - e) * __builtin_amdgcn_rcpf(1.0f + e);
}

__device__ __forceinline__ v8f wmma_f32_k4(v2f a, v2f b, v8f c) {
  // D = A(16x4 f32) * B(4x16 f32) + C(16x16 f32)
  return __builtin_amdgcn_wmma_f32_16x16x4_f32(false, a, false, b, (short)0, c,
                                               false, false);
}

// Issue one 2-D TDM tile load: tile_rows rows of tile_w_elems f32, global row
// stride row_stride_elems, into LDS at lds_off. Descriptor packed per
// CDNA5 ISA (D# group0/group1; groups 2/3 zero -> 2D).
__device__ __forceinline__ void tdm_load_2d(uint32_t lds_off, uint64_t gaddr,
                                            uint32_t tile_w_elems,
                                            uint32_t tile_rows,
                                            uint32_t row_stride_elems) {
  u32x4 g0;
  g0.x = 1u;                                                    // count=1 (valid)
  g0.y = lds_off;                                               // lds_addr (bytes)
  g0.z = (uint32_t)gaddr;                                       // global_addr[31:0]
  g0.w = ((uint32_t)(gaddr >> 32) & 0x01FFFFFFu) | (2u << 30);  // addr[56:32] | type=2
  i32x8 g1;
  g1[0] = (int)(2u << 16);                        // wg_mask=0, data_size=4B
  g1[1] = 0;                                      // tensor_dim0[15:0] (dim0=2^30)
  g1[2] = (int)0x4000u;                           // dim0[31:16] | dim1[15:0]
  g1[3] = (int)(0x4000u | (tile_w_elems << 16));  // dim1[31:16] | tile_dim0
  g1[4] = (int)(tile_rows & 0xFFFFu);             // tile_dim1, tile_dim2=0
  g1[5] = (int)row_stride_elems;                  // tensor_dim0_stride[31:0]
  g1[6] = 0;                                      // stride0[47:32] | stride1[15:0]
  g1[7] = 0;                                      // stride1[47:16]
  i32x4 gz4; gz4[0] = gz4[1] = gz4[2] = gz4[3] = 0;
  i32x8 gz8;
#pragma unroll
  for (int i = 0; i < 8; ++i) gz8[i] = 0;
  __builtin_amdgcn_tensor_load_to_lds(g0, g1, gz4, gz4, gz8, 0);
}

// One LSTM timestep for a 16-batch half. Gates oriented M=gates, N=batch.
// xr points at this lane's x row slice in LDS (the 8 inputs of (batch,t)).
__device__ __forceinline__ void lstm_step(const float* xr, int hi, int lo,
                                          const v2f (&Ax)[2][2],
                                          const v2f (&Ah)[2][2],
                                          const v8f (&biasC)[2],
                                          float (&h)[HD], float (&c)[HD]) {
  // B tiles (K x 16 batch): VGPR v, lane l -> K = 2*(l>=16) + v (+4 for 2nd half)
  v2f Bx0 = { xr[2 * hi + 0], xr[2 * hi + 1] };
  v2f Bx1 = { xr[4 + 2 * hi + 0], xr[4 + 2 * hi + 1] };
  // h^T tile: broadcast each batch's h vector to its column lanes
  float hv[HD];
#pragma unroll
  for (int j = 0; j < HD; ++j) hv[j] = __shfl(h[j], lo, 32);
  v2f Bh0 = { hi ? hv[2] : hv[0], hi ? hv[3] : hv[1] };
  v2f Bh1 = { hi ? 0.0f : hv[4], hi ? 0.0f : hv[5] };  // K=6,7 are zero pad

  v8f c0 = biasC[0];  // gates 0..15
  v8f c1 = biasC[1];  // gates 16..23 (+pad)
  c0 = wmma_f32_k4(Ax[0][0], Bx0, c0);
  c0 = wmma_f32_k4(Ax[0][1], Bx1, c0);
  c0 = wmma_f32_k4(Ah[0][0], Bh0, c0);
  c0 = wmma_f32_k4(Ah[0][1], Bh1, c0);
  c1 = wmma_f32_k4(Ax[1][0], Bx0, c1);
  c1 = wmma_f32_k4(Ax[1][1], Bx1, c1);
  c1 = wmma_f32_k4(Ah[1][0], Bh0, c1);
  c1 = wmma_f32_k4(Ah[1][1], Bh1, c1);

  // C layout: lane l<16 holds gate rows r (c0) / 16+r (c1) of batch l;
  // lane l+16 holds gate rows 8+r of batch l. Pull those 8 across.
  float xch[8];
#pragma unroll
  for (int r = 0; r < 8; ++r) xch[r] = __shfl_xor(c0[r], 16, 32);

  // gate order: i=0..5, f=6..11, g=12..17, o=18..23
  float iv[HD] = { c0[0], c0[1], c0[2], c0[3], c0[4], c0[5] };
  float fv[HD] = { c0[6], c0[7], xch[0], xch[1], xch[2], xch[3] };
  float gv[HD] = { xch[4], xch[5], xch[6], xch[7], c1[0], c1[1] };
  float ov[HD] = { c1[2], c1[3], c1[4], c1[5], c1[6], c1[7] };
#pragma unroll
  for (int j = 0; j < HD; ++j) {
    float ii = fast_sigmoid(iv[j]);
    float ff = fast_sigmoid(fv[j]);
    float gg = fast_tanh(gv[j]);
    float oo = fast_sigmoid(ov[j]);
    c[j] = ff * c[j] + ii * gg;       // valid on lanes 0..15 (batch owners)
    h[j] = oo * fast_tanh(c[j]);
  }
}

__device__ __forceinline__ void write_out(float* out, int Bn, int b,
                                          const float (&h)[HD],
                                          const float* fc_w, const float* fc_b) {
  float acc = fc_b[0];
#pragma unroll
  for (int j = 0; j < HD; ++j) {
    out[Bn + b * HD + j] = h[j];      // feat (second tuple element)
    acc += fc_w[j] * h[j];
  }
  out[b] = fast_sigmoid(acc);         // out (first tuple element)
}

__global__ __launch_bounds__(32) void lstm_fused_kernel(
    const float* __restrict__ x, const float* __restrict__ w_ih,
    const float* __restrict__ w_hh, const float* __restrict__ b_ih,
    const float* __restrict__ b_hh, const float* __restrict__ fc_w,
    const float* __restrict__ fc_b, float* __restrict__ out, int Bn, int T) {
  __shared__ float xs[2][32][TC * IN];  // double-buffered x tiles (2 x 16 KB)

  const int lane = threadIdx.x;
  const int lo = lane & 15;
  const int hi = lane >> 4;
  const int b0 = blockIdx.x * 32;

  // Preload constant A tiles (weights, M=gate rows) and bias C tiles.
  // A 16x4 f32 layout: lane -> M=lo, VGPR v -> K = kh*4 + 2*hi + v.
  v2f Ax[2][2], Ah[2][2];
  v8f biasC[2];
#pragma unroll
  for (int gt = 0; gt < 2; ++gt) {
    int M = gt * 16 + lo;
#pragma unroll
    for (int kh = 0; kh < 2; ++kh) {
#pragma unroll
      for (int v = 0; v < 2; ++v) {
        int K = kh * 4 + 2 * hi + v;
        Ax[gt][kh][v] = (M < NG) ? w_ih[M * IN + K] : 0.0f;
        Ah[gt][kh][v] = (M < NG && K < HD) ? w_hh[M * HD + K] : 0.0f;
      }
    }
#pragma unroll
    for (int r = 0; r < 8; ++r) {  // C layout: VGPR r -> M = gt*16 + 8*hi + r
      int Mg = gt * 16 + 8 * hi + r;
      biasC[gt][r] = (Mg < NG) ? (b_ih[Mg] + b_hh[Mg]) : 0.0f;
    }
  }

  float hA[HD], cA[HD], hB[HD], cB[HD];
#pragma unroll
  for (int j = 0; j < HD; ++j) { hA[j] = cA[j] = hB[j] = cB[j] = 0.0f; }

  const uint64_t xbase =
      (uint64_t)(uintptr_t)x + (uint64_t)b0 * (uint64_t)T * IN * 4ull;
  const uint32_t strideE = (uint32_t)T * IN;
  const int NC = T / TC;

  // Prime the pipeline: chunk 0 -> buffer 0.
  tdm_load_2d((uint32_t)(uintptr_t)&xs[0][0][0], xbase, TC * IN, 32, strideE);

#pragma unroll 1
  for (int ch = 0; ch < NC; ++ch) {
    if (ch + 1 < NC) {
      // Previous reads of the target buffer are long consumed; make sure LDS
      // traffic is quiesced, then prefetch next chunk and wait for current.
      asm volatile("s_wait_dscnt 0" ::: "memory");
      tdm_load_2d((uint32_t)(uintptr_t)&xs[(ch + 1) & 1][0][0],
                  xbase + (uint64_t)(ch + 1) * TC * IN * 4ull, TC * IN, 32,
                  strideE);
      __builtin_amdgcn_s_wait_tensorcnt(1);  // TDM in-order: chunk ch done
    } else {
      __builtin_amdgcn_s_wait_tensorcnt(0);
    }
    const float(*xt)[TC * IN] = xs[ch & 1];
#pragma unroll 1
    for (int tr = 0; tr < TC; ++tr) {
      lstm_step(&xt[lo][tr * IN],      hi, lo, Ax, Ah, biasC, hA, cA);
      lstm_step(&xt[16 + lo][tr * IN], hi, lo, Ax, Ah, biasC, hB, cB);
    }
  }

  if (lane < 16) {  // lanes 0..15 own the final h of both batch halves
    write_out(out, Bn, b0 + lane, hA, fc_w, fc_b);
    write_out(out, Bn, b0 + 16 + lane, hB, fc_w, fc_b);
  }
}

extern "C" void kernel_launch(void* const* d_in, const int* in_sizes, int n_in,
                              void* d_out, int out_size, void* d_ws,
                              size_t ws_size, hipStream_t stream) {
  const float* x    = (const float*)d_in[0];
  const float* w_ih = (const float*)d_in[1];
  const float* w_hh = (const float*)d_in[2];
  const float* b_ih = (const float*)d_in[3];
  const float* b_hh = (const float*)d_in[4];
  const float* fc_w = (const float*)d_in[5];
  const float* fc_b = (const float*)d_in[6];
  const int Bn = out_size / 7;            // out [B,1] ++ feat [B,6]
  const int T  = in_sizes[0] / (Bn * 8);  // x is [B,T,8]
  dim3 grid(Bn / 32), block(32);
  hipLaunchKernelGGL(lstm_fused_kernel, grid, block, 0, stream, x, w_ih, w_hh,
                     b_ih, b_hh, fc_w, fc_b, (float*)d_out, Bn, T);
}